// GATv2Layer_46188078301403
// MI455X (gfx1250) — compile-verified
//
#include <hip/hip_runtime.h>
#include <hip/hip_bf16.h>
#include <math.h>

// Problem constants from the reference
#define NN 1024   // nodes
#define CC 256    // channels
#define NEG_SLOPE 0.2f

typedef __attribute__((ext_vector_type(2))) float v2f;
typedef __attribute__((ext_vector_type(8))) float v8f;

// ---------------------------------------------------------------------------
// K1: g = nodes @ W.T + b   (two GEMMs selected by blockIdx.y)
// One wave32 per 16x16 tile of the [1024,256] output, K-loop of 64
// v_wmma_f32_16x16x4_f32 ops.
//
// Fragment layouts per CDNA5 ISA 7.12.2 (f32):
//   A 16x4:  lanes 0-15 -> rows M=0..15, VGPR0=K+0, VGPR1=K+1
//            lanes 16-31 -> rows M=0..15, VGPR0=K+2, VGPR1=K+3
//   B 4x16:  mirrored (row K striped across lanes within a VGPR)
//   C/D:     VGPR v: lanes 0-15 -> M=v, N=lane; lanes 16-31 -> M=v+8, N=lane-16
// ---------------------------------------------------------------------------
__global__ void __launch_bounds__(128)
gat_proj_gemm(const float* __restrict__ nodes,
              const float* __restrict__ Wsrc, const float* __restrict__ bsrc,
              const float* __restrict__ Wtgt, const float* __restrict__ btgt,
              float* __restrict__ gsrc, float* __restrict__ gtgt)
{
    const int wave = threadIdx.x >> 5;
    const int lane = threadIdx.x & 31;
    const int half = lane >> 4;        // 0: lanes 0-15, 1: lanes 16-31
    const int m    = lane & 15;

    const int tile = blockIdx.x * 4 + wave;   // 0..1023 tiles of [1024/16, 256/16]
    const int tm   = tile >> 4;               // row tile
    const int tn   = tile & 15;               // col tile

    const bool is_src = (blockIdx.y == 0);
    const float* __restrict__ W    = is_src ? Wsrc : Wtgt;   // [C,C] row-major
    const float* __restrict__ bias = is_src ? bsrc : btgt;   // [C]
    float* __restrict__ out        = is_src ? gsrc : gtgt;   // [N,C]

    const int row = tm * 16 + m;   // A-matrix row this lane feeds
    const int col = tn * 16 + m;   // B/D column this lane feeds

    v8f acc = {};
    #pragma unroll 4
    for (int k = 0; k < CC; k += 4) {
        const int kk = k + 2 * half;
        // A[row][kk], A[row][kk+1] : contiguous -> float2 load
        v2f a = *(const v2f*)(nodes + row * CC + kk);
        // B[kk][col] = W[col][kk]  (B = W.T) : contiguous in W row -> float2 load
        v2f b = *(const v2f*)(W + col * CC + kk);
        acc = __builtin_amdgcn_wmma_f32_16x16x4_f32(
            /*neg_a=*/false, a, /*neg_b=*/false, b,
            /*c_mod=*/(short)0, acc, /*reuse_a=*/false, /*reuse_b=*/false);
    }

    const float bn = bias[col];   // bias depends on column only
    #pragma unroll
    for (int v = 0; v < 8; ++v) {
        out[(tm * 16 + v + 8 * half) * CC + col] = acc[v] + bn;
    }
}

// ---------------------------------------------------------------------------
// K2: per row i: score[j] = sum_c a[c]*lrelu(gs[i,c]+gt[j,c]); mask by adj;
// softmax over j; write weights row. One block (256 threads) per i, each
// thread owns 4 j's. gs[i,:] and a_w staged in LDS; gt rows are L2-resident.
// ---------------------------------------------------------------------------
__global__ void __launch_bounds__(256)
gat_score_softmax(const float* __restrict__ gs,
                  const float* __restrict__ gt,
                  const float* __restrict__ aw,
                  const int*   __restrict__ adj,
                  float* __restrict__ wts)
{
    __shared__ float sh_gs[CC];
    __shared__ float sh_a[CC];
    __shared__ float red[256];

    const int i = blockIdx.x;
    const int t = threadIdx.x;

    sh_gs[t] = gs[i * CC + t];
    sh_a[t]  = aw[t];
    __syncthreads();

    float s[4];
    #pragma unroll
    for (int q = 0; q < 4; ++q) {
        const int j = t + q * 256;
        const float4* __restrict__ gtj = (const float4*)(gt + j * CC);
        if (q < 3) __builtin_prefetch(gt + (j + 256) * CC, 0, 1);
        float acc = 0.f;
        #pragma unroll 4
        for (int c = 0; c < CC; c += 4) {
            float4 g4 = gtj[c >> 2];
            float x0 = sh_gs[c + 0] + g4.x;
            float x1 = sh_gs[c + 1] + g4.y;
            float x2 = sh_gs[c + 2] + g4.z;
            float x3 = sh_gs[c + 3] + g4.w;
            // leaky_relu(x) = max(x,0) + slope*min(x,0)
            acc += sh_a[c + 0] * (fmaxf(x0, 0.f) + NEG_SLOPE * fminf(x0, 0.f));
            acc += sh_a[c + 1] * (fmaxf(x1, 0.f) + NEG_SLOPE * fminf(x1, 0.f));
            acc += sh_a[c + 2] * (fmaxf(x2, 0.f) + NEG_SLOPE * fminf(x2, 0.f));
            acc += sh_a[c + 3] * (fmaxf(x3, 0.f) + NEG_SLOPE * fminf(x3, 0.f));
        }
        s[q] = (adj[i * NN + j] != 0) ? acc : -INFINITY;
    }

    // block-wide max over the 1024 scores
    float mx = fmaxf(fmaxf(s[0], s[1]), fmaxf(s[2], s[3]));
    red[t] = mx;
    __syncthreads();
    for (int off = 128; off > 0; off >>= 1) {
        if (t < off) red[t] = fmaxf(red[t], red[t + off]);
        __syncthreads();
    }
    const float mxAll = red[0];
    __syncthreads();

    // exp + block-wide sum (masked entries contribute exactly 0)
    float e[4];
    float lsum = 0.f;
    #pragma unroll
    for (int q = 0; q < 4; ++q) {
        e[q] = (s[q] > -INFINITY) ? __expf(s[q] - mxAll) : 0.f;
        lsum += e[q];
    }
    red[t] = lsum;
    __syncthreads();
    for (int off = 128; off > 0; off >>= 1) {
        if (t < off) red[t] += red[t + off];
        __syncthreads();
    }
    const float inv = 1.f / red[0];

    #pragma unroll
    for (int q = 0; q < 4; ++q)
        wts[i * NN + t + q * 256] = e[q] * inv;
}

// ---------------------------------------------------------------------------
// K3: out = weights @ g_tgt  ([1024x1024] x [1024x256]) with f32 WMMA,
// K = 1024 -> 256 wmma ops per 16x16 tile.
// ---------------------------------------------------------------------------
__global__ void __launch_bounds__(128)
gat_out_gemm(const float* __restrict__ wts,   // [N,N]
             const float* __restrict__ gt,    // [N,C]
             float* __restrict__ out)         // [N,C]
{
    const int wave = threadIdx.x >> 5;
    const int lane = threadIdx.x & 31;
    const int half = lane >> 4;
    const int m    = lane & 15;

    const int tile = blockIdx.x * 4 + wave;   // 0..1023
    const int tm   = tile >> 4;
    const int tn   = tile & 15;

    const int row = tm * 16 + m;
    const int col = tn * 16 + m;

    v8f acc = {};
    #pragma unroll 2
    for (int k = 0; k < NN; k += 4) {
        const int kk = k + 2 * half;
        // A[row][kk..kk+1] contiguous
        v2f a = *(const v2f*)(wts + row * NN + kk);
        // B[kk][col] = gt[kk][col] (row-major, stride C) -> two strided loads
        v2f b;
        b.x = gt[kk * CC + col];
        b.y = gt[(kk + 1) * CC + col];
        acc = __builtin_amdgcn_wmma_f32_16x16x4_f32(
            false, a, false, b, (short)0, acc, false, false);
    }

    #pragma unroll
    for (int v = 0; v < 8; ++v) {
        out[(tm * 16 + v + 8 * half) * CC + col] = acc[v];
    }
}

// ---------------------------------------------------------------------------
// Host-side launch. Input order per setup_inputs():
//   0: nodes [N,C] f32      1: adj_mat [N,N] i32
//   2: W_src_w [C,C] f32    3: W_src_b [C] f32
//   4: W_tgt_w [C,C] f32    5: W_tgt_b [C] f32
//   6: a_w [C] f32
// Output: [N,C] f32. Workspace layout: gsrc(1MB) | gtgt(1MB) | weights(4MB).
// ---------------------------------------------------------------------------
extern "C" void kernel_launch(void* const* d_in, const int* in_sizes, int n_in,
                              void* d_out, int out_size, void* d_ws, size_t ws_size,
                              hipStream_t stream)
{
    const float* nodes = (const float*)d_in[0];
    const int*   adj   = (const int*)  d_in[1];
    const float* Wsrc  = (const float*)d_in[2];
    const float* bsrc  = (const float*)d_in[3];
    const float* Wtgt  = (const float*)d_in[4];
    const float* btgt  = (const float*)d_in[5];
    const float* aw    = (const float*)d_in[6];
    float* out = (float*)d_out;

    char* ws = (char*)d_ws;
    float* gsrc = (float*)(ws);
    float* gtgt = (float*)(ws + (size_t)(1 << 20));
    float* wts  = (float*)(ws + (size_t)(2 << 20));

    // K1: both projections (y=0 -> src, y=1 -> tgt); 256 blocks x 4 waves = 1024 tiles
    gat_proj_gemm<<<dim3(256, 2), 128, 0, stream>>>(nodes, Wsrc, bsrc, Wtgt, btgt,
                                                    gsrc, gtgt);
    // K2: one block per target row i
    gat_score_softmax<<<NN, 256, 0, stream>>>(gsrc, gtgt, aw, adj, wts);
    // K3: output GEMM, 1024 tiles
    gat_out_gemm<<<256, 128, 0, stream>>>(wts, gtgt, out);
}